// EfficientGNN_36378372997643
// MI455X (gfx1250) — compile-verified
//
#include <hip/hip_runtime.h>
#include <hip/hip_bf16.h>
#include <stdint.h>

typedef __attribute__((ext_vector_type(2))) float v2f;
typedef __attribute__((ext_vector_type(8))) float v8f;

#if __has_builtin(__builtin_amdgcn_global_load_async_to_lds_b128)
#define HAVE_ASYNC_LDS 1
typedef int v4i __attribute__((vector_size(16)));
typedef __attribute__((address_space(1))) v4i* gv4i_p;   // global (AS1) v4i*
typedef __attribute__((address_space(3))) v4i* lv4i_p;   // LDS    (AS3) v4i*
#else
#define HAVE_ASYNC_LDS 0
#endif

// ---------------------------------------------------------------------------
// fill buffer with constant (grid-stride)
// ---------------------------------------------------------------------------
__global__ void fill_f32(float* __restrict__ p, float v, long n) {
    long i = (long)blockIdx.x * blockDim.x + threadIdx.x;
    long stride = (long)gridDim.x * blockDim.x;
    for (; i < n; i += stride) p[i] = v;
}

// ---------------------------------------------------------------------------
// degree count: deg[col[e]] += 1  (deg pre-filled with 1.0 for self loops)
// ---------------------------------------------------------------------------
__global__ void deg_count(const int* __restrict__ col, float* __restrict__ deg, int E) {
    int e = blockIdx.x * blockDim.x + threadIdx.x;
    if (e < E) atomicAdd(&deg[col[e]], 1.0f);
}

// deg -> deg^{-1/2} in place
__global__ void deg_to_dis(float* __restrict__ deg, int N) {
    int i = blockIdx.x * blockDim.x + threadIdx.x;
    if (i < N) {
        float d = deg[i];
        deg[i] = (d > 0.0f) ? rsqrtf(d) : 0.0f;
    }
}

// ---------------------------------------------------------------------------
// H[N x 128] = X[N x 128] @ W[128 x 128] using V_WMMA_F32_16X16X4_F32.
//
// Block = 256 threads = 8 waves; block handles 32 rows x 128 cols.
// The 32x128 A tile (16 KB) is staged into LDS via async global->LDS copy.
// Wave w owns columns [16w, 16w+16) and computes TWO 16x16 C tiles (rows
// m0..m0+15 and m0+16..m0+31), so each B load feeds two WMMAs.
//
// fp32 WMMA lane layout (ISA 7.12.2):
//   A 16x4 : lanes 0-15: M=lane, v0=K0,v1=K1 ; lanes 16-31: M=lane-16, v0=K2,v1=K3
//   B 4x16 : lanes 0-15: N=lane, v0=K0,v1=K1 ; lanes 16-31: N=lane-16, v0=K2,v1=K3
//   C 16x16: VGPR j: lanes 0-15 -> M=j, lanes 16-31 -> M=8+j ; N=lane&15
// ---------------------------------------------------------------------------
__global__ __launch_bounds__(256) void gemm_xw_wmma(const float* __restrict__ X,
                                                    const float* __restrict__ W,
                                                    float* __restrict__ H, int N) {
    __shared__ float As[32 * 128];   // 16 KB

    const int tid  = threadIdx.x;
    const int wave = tid >> 5;
    const int lane = tid & 31;
    const int m0   = blockIdx.x * 32;
    const int n0   = wave * 16;
    const int r    = lane & 15;          // M (for A) / N (for B)
    const int kh   = (lane >> 4) * 2;    // 0 or 2: which K-half this lane holds

    // --- cooperative stage of A tile: 1024 float4 chunks, 4 per thread ----
    for (int i = tid; i < 1024; i += 256) {
        const int rrow = i >> 5;              // row within tile, 0..31
        const int cc   = (i & 31) * 4;        // float offset within row
        int grow = m0 + rrow;
        if (grow >= N) grow = N - 1;          // clamp (N % 32 == 0 anyway)
        const float* src = X + (size_t)grow * 128 + cc;
        float*       dst = &As[rrow * 128 + cc];
#if HAVE_ASYNC_LDS
        __builtin_amdgcn_global_load_async_to_lds_b128(
            (gv4i_p)(v4i*)src, (lv4i_p)(v4i*)dst, 0, 0);
#else
        *(float4*)dst = *(const float4*)src;
#endif
    }
#if HAVE_ASYNC_LDS
    asm volatile("s_wait_asynccnt 0x0" ::: "memory");
#endif
    __syncthreads();

    // --- K loop: 32 steps of K=4, 2 WMMAs per step ------------------------
    v8f c0 = {}, c1 = {};
    const float* __restrict__ wcol = W + n0 + r;
#pragma unroll 4
    for (int k = 0; k < 128; k += 4) {
        v2f a0, a1, b;
        a0.x = As[r * 128 + k + kh];
        a0.y = As[r * 128 + k + kh + 1];
        a1.x = As[(r + 16) * 128 + k + kh];
        a1.y = As[(r + 16) * 128 + k + kh + 1];
        b.x  = wcol[(size_t)(k + kh) * 128];
        b.y  = wcol[(size_t)(k + kh + 1) * 128];
        c0 = __builtin_amdgcn_wmma_f32_16x16x4_f32(false, a0, false, b,
                                                   (short)0, c0, false, false);
        c1 = __builtin_amdgcn_wmma_f32_16x16x4_f32(false, a1, false, b,
                                                   (short)0, c1, false, false);
    }

    // --- store two 16x16 tiles -------------------------------------------
    const int col = n0 + r;
    const int rb0 = m0 + ((lane >> 4) * 8);
    const int rb1 = rb0 + 16;
    if (m0 + 32 <= N) {               // fast path: fully in-bounds, no EXEC games
        float* h0 = H + (size_t)rb0 * 128 + col;
        float* h1 = H + (size_t)rb1 * 128 + col;
#pragma unroll
        for (int j = 0; j < 8; ++j) {
            h0[(size_t)j * 128] = c0[j];
            h1[(size_t)j * 128] = c1[j];
        }
    } else {
#pragma unroll
        for (int j = 0; j < 8; ++j) {
            if (rb0 + j < N) H[(size_t)(rb0 + j) * 128 + col] = c0[j];
            if (rb1 + j < N) H[(size_t)(rb1 + j) * 128 + col] = c1[j];
        }
    }
}

// ---------------------------------------------------------------------------
// edge scatter: A[c, :] += H[r, :] * dis[r] * dis[c]
// one wave per edge; lane handles 4 contiguous feats (float4 load, 4 atomics)
// ---------------------------------------------------------------------------
__global__ void scatter_edges(const int* __restrict__ row, const int* __restrict__ col,
                              const float* __restrict__ dis, const float* __restrict__ H,
                              float* __restrict__ A, int E) {
    int e = blockIdx.x * 8 + (threadIdx.x >> 5);
    if (e >= E) return;
    const int lane = threadIdx.x & 31;
    const int r = row[e];
    const int c = col[e];
    const float nrm = dis[r] * dis[c];
    const float4 v = ((const float4*)(H + (size_t)r * 128))[lane];
    float* dst = A + (size_t)c * 128 + lane * 4;
    atomicAdd(dst + 0, v.x * nrm);
    atomicAdd(dst + 1, v.y * nrm);
    atomicAdd(dst + 2, v.z * nrm);
    atomicAdd(dst + 3, v.w * nrm);
}

// ---------------------------------------------------------------------------
// finalize: A[v,:] = relu(A[v,:] + H[v,:] * dis[v]^2 + b)  (self-loop + bias)
// ---------------------------------------------------------------------------
__global__ void finalize_relu(float* __restrict__ A, const float* __restrict__ H,
                              const float* __restrict__ dis, const float* __restrict__ b,
                              int N) {
    long idx = (long)blockIdx.x * blockDim.x + threadIdx.x;
    long total = (long)N * 32;
    if (idx >= total) return;
    const int v  = (int)(idx >> 5);
    const int f4 = (int)(idx & 31);
    const float s = dis[v];
    const float s2 = s * s;
    float4 a = ((float4*)A)[idx];
    const float4 h = ((const float4*)H)[idx];
    const float4 bb = ((const float4*)b)[f4];
    a.x = fmaxf(fmaf(h.x, s2, a.x) + bb.x, 0.0f);
    a.y = fmaxf(fmaf(h.y, s2, a.y) + bb.y, 0.0f);
    a.z = fmaxf(fmaf(h.z, s2, a.z) + bb.z, 0.0f);
    a.w = fmaxf(fmaf(h.w, s2, a.w) + bb.w, 0.0f);
    ((float4*)A)[idx] = a;
}

// ---------------------------------------------------------------------------
// pooling over sorted batch ids: pooled[g,:] += H[v,:]
// block = 128 threads (one per feature), each walks 512 nodes, register
// run-length accumulation, one atomic flush per (graph-run, feature).
// ---------------------------------------------------------------------------
__global__ void pool_nodes(const float* __restrict__ H, const int* __restrict__ batch,
                           float* __restrict__ pooled, int N) {
    const int f  = threadIdx.x;
    const int n0 = blockIdx.x * 512;
    if (n0 >= N) return;
    const int n1 = min(n0 + 512, N);
    int g = batch[n0];
    float acc = 0.0f;
    for (int n = n0; n < n1; ++n) {
        const int gn = batch[n];
        if (gn != g) {
            atomicAdd(&pooled[g * 128 + f], acc);
            acc = 0.0f;
            g = gn;
        }
        acc += H[(size_t)n * 128 + f];
    }
    atomicAdd(&pooled[g * 128 + f], acc);
}

// ---------------------------------------------------------------------------
// out[128 x 2] = pooled[128 x 128] @ Wl[128 x 2] + bl  (tiny, one block)
// ---------------------------------------------------------------------------
__global__ void final_linear(const float* __restrict__ pooled, const float* __restrict__ Wl,
                             const float* __restrict__ bl, float* __restrict__ out) {
    const int t = threadIdx.x;
    if (t >= 256) return;
    const int g = t >> 1;
    const int c = t & 1;
    float acc = bl[c];
#pragma unroll 16
    for (int k = 0; k < 128; ++k)
        acc = fmaf(pooled[g * 128 + k], Wl[k * 2 + c], acc);
    out[t] = acc;
}

// ---------------------------------------------------------------------------
extern "C" void kernel_launch(void* const* d_in, const int* in_sizes, int n_in,
                              void* d_out, int out_size, void* d_ws, size_t ws_size,
                              hipStream_t stream) {
    const float* x    = (const float*)d_in[0];
    const int*   ei   = (const int*)d_in[1];   // [2, E]
    const int*   bat  = (const int*)d_in[2];   // [N]
    const float* W1   = (const float*)d_in[3];
    const float* b1   = (const float*)d_in[4];
    const float* W2   = (const float*)d_in[5];
    const float* b2   = (const float*)d_in[6];
    const float* Wl   = (const float*)d_in[7];
    const float* bl   = (const float*)d_in[8];
    float*       out  = (float*)d_out;

    const int N = in_sizes[0] / 128;
    const int E = in_sizes[1] / 2;
    const int* row = ei;
    const int* col = ei + E;

    // workspace layout (256B aligned)
    char* ws = (char*)d_ws;
    size_t off = 0;
    auto alloc = [&](size_t bytes) {
        void* p = ws + off;
        off += (bytes + 255) & ~(size_t)255;
        return p;
    };
    float* deg    = (float*)alloc((size_t)N * sizeof(float));          // deg -> dis
    float* bufA   = (float*)alloc((size_t)N * 128 * sizeof(float));    // H (XW)
    float* bufB   = (float*)alloc((size_t)N * 128 * sizeof(float));    // aggregated
    float* pooled = (float*)alloc((size_t)128 * 128 * sizeof(float));
    (void)ws_size; (void)n_in; (void)out_size;

    const long NF = (long)N * 128;
    const int  gemmBlocks = (N + 31) / 32;

    // --- degree / normalization ---------------------------------------
    fill_f32<<<2048, 256, 0, stream>>>(deg, 1.0f, (long)N);            // self loop
    deg_count<<<(E + 255) / 256, 256, 0, stream>>>(col, deg, E);
    deg_to_dis<<<(N + 255) / 256, 256, 0, stream>>>(deg, N);

    // --- layer 1: H = X @ W1 ; A = scatter ; relu(A + self + b1) ------
    gemm_xw_wmma<<<gemmBlocks, 256, 0, stream>>>(x, W1, bufA, N);
    fill_f32<<<4096, 256, 0, stream>>>(bufB, 0.0f, NF);
    scatter_edges<<<(E + 7) / 8, 256, 0, stream>>>(row, col, deg, bufA, bufB, E);
    finalize_relu<<<(int)(((long)N * 32 + 255) / 256), 256, 0, stream>>>(bufB, bufA, deg, b1, N);

    // --- layer 2: H2 = A1 @ W2 ; A2 = scatter ; relu(...) -------------
    gemm_xw_wmma<<<gemmBlocks, 256, 0, stream>>>(bufB, W2, bufA, N);
    fill_f32<<<4096, 256, 0, stream>>>(bufB, 0.0f, NF);
    scatter_edges<<<(E + 7) / 8, 256, 0, stream>>>(row, col, deg, bufA, bufB, E);
    finalize_relu<<<(int)(((long)N * 32 + 255) / 256), 256, 0, stream>>>(bufB, bufA, deg, b2, N);

    // --- pooling + classifier -----------------------------------------
    fill_f32<<<64, 256, 0, stream>>>(pooled, 0.0f, 128L * 128L);
    pool_nodes<<<(N + 511) / 512, 128, 0, stream>>>(bufB, bat, pooled, N);
    final_linear<<<1, 256, 0, stream>>>(pooled, Wl, bl, out);
}